// Fastfood_46703474377367
// MI455X (gfx1250) — compile-verified
//
#include <hip/hip_runtime.h>

// Fastfood random features for MI455X (gfx1250), fp32 end-to-end.
// FWHT_1024 = H16 (x) H16 (x) H4 over index j = a*64 + b*4 + c.
//   - H4 (axis c, low 2 bits): done in registers, fused into the global-load
//     phase (FWHT#1) and the permutation write phase (FWHT#2) -- zero LDS cost.
//   - H16 (axes b and a): V_WMMA_F32_16X16X4_F32, 4 chained K-chunks per
//     16(batch rows) x 16 tile, in-place in LDS (groups are wave-disjoint).
//
// One workgroup (256 thr = 8 wave32) owns a 16-row x 1024-col fp32 tile in
// 64 KB LDS. XOR swizzle (j ^ (r<<2)) gives conflict-free column access while
// keeping 16B quads contiguous for b128 LDS ops.

typedef float v2f __attribute__((ext_vector_type(2)));
typedef float v8f __attribute__((ext_vector_type(8)));

#define DD 1024
// swizzled LDS index for (row r in [0,16), col j in [0,1024))
#define SW(r, j) ((((r)) << 10) | ((j) ^ (((r) << 2) & 60)))

// 4-point Hadamard on a register quad (Sylvester order).
__device__ __forceinline__ float4 h4(float4 v) {
  const float s0 = v.x + v.y, s1 = v.x - v.y;
  const float s2 = v.z + v.w, s3 = v.z - v.w;
  return make_float4(s0 + s2, s1 + s3, s0 - s2, s1 - s3);
}

// H16 transform along one axis (stride 4 = axis b, stride 64 = axis a) over
// 64 groups split across 8 waves. Data tile: lds[SW(r, base + k*stride)],
// r = batch row (M of WMMA), k = transformed index.
// D = A(16x4 chunks of T) x B(4x16 chunks of H16), accumulated over 4 chunks.
// H16 is symmetric so left/right multiply agree.
__device__ __forceinline__ void h16_stage(float* lds, int lane, int wid,
                                          const v2f Hb[4], int stride_is_64) {
  const int hi = lane >> 4;        // lane half: selects K pair (A) / rows (B,D)
  const int m  = lane & 15;        // A-operand row = batch row
  const int nn = lane & 15;        // D-operand column
  for (int it = 0; it < 8; ++it) {
    const int g = wid * 8 + it;    // group id 0..63
    // stride 4  (axis b): base = a*64 + c, a=g>>2, c=g&3
    // stride 64 (axis a): base = b*4  + c  = g
    const int base   = stride_is_64 ? g : (((g >> 2) << 6) | (g & 3));
    const int stride = stride_is_64 ? 64 : 4;
    v8f acc = {0.f, 0.f, 0.f, 0.f, 0.f, 0.f, 0.f, 0.f};
#pragma unroll
    for (int q = 0; q < 4; ++q) {
      const int k0 = 4 * q + 2 * hi;   // absolute K of this lane's A pair
      v2f a;
      a[0] = lds[SW(m, base + (k0 + 0) * stride)];
      a[1] = lds[SW(m, base + (k0 + 1) * stride)];
      // (neg_a, A, neg_b, B, c_mod, C, reuse_a, reuse_b)
      acc = __builtin_amdgcn_wmma_f32_16x16x4_f32(
          false, a, false, Hb[q], (short)0, acc, false, false);
    }
    // D layout: VGPR p, lane l -> D[p + 8*(l>>4)][l&15]
#pragma unroll
    for (int p = 0; p < 8; ++p)
      lds[SW(p + 8 * hi, base + nn * stride)] = acc[p];
  }
}

// The two WMMA H16 stages of an FWHT (caller has already applied H4 along c).
__device__ __forceinline__ void fwht_h16(float* lds, int lane, int wid,
                                         const v2f Hb[4]) {
  h16_stage(lds, lane, wid, Hb, /*stride_is_64=*/0);   // axis b
  __syncthreads();
  h16_stage(lds, lane, wid, Hb, /*stride_is_64=*/1);   // axis a
  __syncthreads();
}

__global__ __launch_bounds__(256) void fastfood_kernel(
    const float* __restrict__ x, const float* __restrict__ Bm,
    const int* __restrict__ Pi, const float* __restrict__ G,
    const float* __restrict__ S, float* __restrict__ out) {
  __shared__ float lds[16 * DD];   // 64 KB

  const int t    = threadIdx.x;    // 0..255
  const int lane = t & 31;
  const int wid  = t >> 5;
  const int bq   = blockIdx.y;     // fastfood block 0..3
  const int row0 = blockIdx.x * 16;

  // Constant B-operand: H16 chunks. B layout (f32 4x16, 2 VGPRs):
  // lanes 0-15 hold K = {0,1} in v{0,1}; lanes 16-31 hold K = {2,3}.
  const int hi = lane >> 4, nn = lane & 15;
  v2f Hb[4];
#pragma unroll
  for (int q = 0; q < 4; ++q) {
    const int k0 = 4 * q + 2 * hi;
    Hb[q][0] = (__builtin_popcount((k0 + 0) & nn) & 1) ? -1.0f : 1.0f;
    Hb[q][1] = (__builtin_popcount((k0 + 1) & nn) & 1) ? -1.0f : 1.0f;
  }

  // ---- load x tile; fuse B[b] sign multiply AND the H4 stage of FWHT#1
  //      (each loaded quad (r, 4t..4t+3) is a complete c-quad).
  {
    const float4* x4 = reinterpret_cast<const float4*>(x) + (size_t)row0 * (DD / 4);
    const float4* B4 = reinterpret_cast<const float4*>(Bm) + (size_t)bq * (DD / 4);
    const float4 bb = B4[t];
#pragma unroll
    for (int r = 0; r < 16; ++r) {
      float4 v = x4[(size_t)r * (DD / 4) + t];
      v.x *= bb.x; v.y *= bb.y; v.z *= bb.z; v.w *= bb.w;
      *reinterpret_cast<float4*>(&lds[SW(r, 4 * t)]) = h4(v);
    }
  }
  __syncthreads();

  // ---- FWHT #1, remaining H16 stages (WMMA, in place)
  fwht_h16(lds, lane, wid, Hb);

  // ---- permutation gather + G multiply, staged through registers so a
  //      single LDS buffer suffices (read-all, barrier, write-all in place).
  //      The H4 stage of FWHT#2 is fused into the write (tmp quads = c-quads).
  {
    const int jb = 4 * t;
    const int4  pi = *reinterpret_cast<const int4*>(Pi + (size_t)bq * DD + jb);
    const float4 g = *reinterpret_cast<const float4*>(G + (size_t)bq * DD + jb);
    float4 tmp[16];
#pragma unroll
    for (int r = 0; r < 16; ++r) {
      tmp[r].x = lds[SW(r, pi.x)] * g.x;
      tmp[r].y = lds[SW(r, pi.y)] * g.y;
      tmp[r].z = lds[SW(r, pi.z)] * g.z;
      tmp[r].w = lds[SW(r, pi.w)] * g.w;
    }
    __syncthreads();
#pragma unroll
    for (int r = 0; r < 16; ++r)
      *reinterpret_cast<float4*>(&lds[SW(r, jb)]) = h4(tmp[r]);
    __syncthreads();
  }

  // ---- FWHT #2, remaining H16 stages (WMMA, in place)
  fwht_h16(lds, lane, wid, Hb);

  // ---- fuse S * (1/(sigma*sqrt(d))) and store (coalesced b128)
  {
    const float4 s = *reinterpret_cast<const float4*>(S + (size_t)bq * DD + 4 * t);
    const float sc = 0.03125f;  // 1/32
#pragma unroll
    for (int r = 0; r < 16; ++r) {
      float4 v = *reinterpret_cast<float4*>(&lds[SW(r, 4 * t)]);
      v.x *= s.x * sc; v.y *= s.y * sc; v.z *= s.z * sc; v.w *= s.w * sc;
      *reinterpret_cast<float4*>(out + (size_t)(row0 + r) * (4 * DD) +
                                 (size_t)bq * DD + 4 * t) = v;
    }
  }
}

extern "C" void kernel_launch(void* const* d_in, const int* in_sizes, int n_in,
                              void* d_out, int out_size, void* d_ws, size_t ws_size,
                              hipStream_t stream) {
  const float* x  = (const float*)d_in[0];
  const float* Bm = (const float*)d_in[1];
  const int*   Pi = (const int*)d_in[2];
  const float* G  = (const float*)d_in[3];
  const float* S  = (const float*)d_in[4];
  float* out = (float*)d_out;

  const int N = in_sizes[0] / DD;       // 16384
  dim3 grid(N / 16, 4);                 // 16-row tiles x 4 fastfood blocks
  fastfood_kernel<<<grid, 256, 0, stream>>>(x, Bm, Pi, G, S, out);
}